// SubPathwayModel_2061584302288
// MI455X (gfx1250) — compile-verified
//
#include <hip/hip_runtime.h>
#include <hip/hip_bf16.h>
#include <stdint.h>

// ---------- constants from the reference ----------
#define P_NODES 30000
#define E_EDGES 480000
#define G_GENES 20000
#define MP_PATH 32
#define R_ROOT  28
#define B_BATCH 4096
#define IN_DIM  394
#define H_DIM   256
#define OUT_DIM 64
#define GD_DIM  128

typedef __attribute__((ext_vector_type(2))) float v2f;
typedef __attribute__((ext_vector_type(8))) float v8f;

// =====================================================================
// FP32 WMMA GEMM: C[M,N] = A[M,K] @ W[K,N] (+bias) (opt relu)
// 4 waves per block; each wave owns one 16(M) x 64(N) tile (4 accumulators
// reuse one A fragment). Main K loop is branch-free; K%4 tail handled once
// with clamped loads + v_cndmask zeroing (no exec divergence on loads).
// Requires: M % 16 == 0, N % 64 == 0 (true for every call here).
// =====================================================================
__global__ __launch_bounds__(128)
void gemm_wmma_f32_kernel(const float* __restrict__ A,
                          const float* __restrict__ W,
                          const float* __restrict__ bias,
                          float* __restrict__ C,
                          int M, int N, int K, int do_relu)
{
    const int wave = threadIdx.x >> 5;
    const int lane = threadIdx.x & 31;
    const int hh   = lane >> 4;    // lane half: 0 -> lanes 0-15, 1 -> 16-31
    const int lr   = lane & 15;    // row (A) / col (B,C) within half

    const int tilesN = N >> 6;
    const int T = (M >> 4) * tilesN;
    const int tile = blockIdx.x * 4 + wave;
    if (tile >= T) return;                      // uniform per wave
    const int row0 = (tile / tilesN) * 16;
    const int col0 = (tile % tilesN) * 64;

    v8f acc0 = {}; v8f acc1 = {}; v8f acc2 = {}; v8f acc3 = {};

    const int K4 = K & ~3;
    const float* __restrict__ aptr = A + (size_t)(row0 + lr) * K + 2 * hh;
    const float* __restrict__ wptr = W + (size_t)(2 * hh) * N + col0 + lr;
    const size_t wstep = (size_t)4 * N;

    #pragma unroll 2
    for (int k0 = 0; k0 < K4; k0 += 4) {
        v2f a;
        a.x = aptr[0];
        a.y = aptr[1];
        const float* w1p = wptr + N;
        v2f b0, b1, b2, b3;
        b0.x = wptr[0];   b0.y = w1p[0];
        b1.x = wptr[16];  b1.y = w1p[16];
        b2.x = wptr[32];  b2.y = w1p[32];
        b3.x = wptr[48];  b3.y = w1p[48];

        acc0 = __builtin_amdgcn_wmma_f32_16x16x4_f32(false, a, false, b0, (short)0, acc0, false, false);
        acc1 = __builtin_amdgcn_wmma_f32_16x16x4_f32(false, a, false, b1, (short)0, acc1, false, false);
        acc2 = __builtin_amdgcn_wmma_f32_16x16x4_f32(false, a, false, b2, (short)0, acc2, false, false);
        acc3 = __builtin_amdgcn_wmma_f32_16x16x4_f32(false, a, false, b3, (short)0, acc3, false, false);

        aptr += 4;
        wptr += wstep;
    }

    if (K4 < K) {
        // tail (only K=394 hits this): clamp addresses, select-zero the
        // out-of-range K contributions -> plain v_cndmask, no branches.
        const int ka  = K4 + 2 * hh;
        const bool ok0 = (ka     < K);
        const bool ok1 = (ka + 1 < K);
        const int kc0 = ok0 ? ka     : (K - 1);
        const int kc1 = ok1 ? ka + 1 : (K - 1);

        const float* Ar = A + (size_t)(row0 + lr) * K;
        v2f a;
        float a0 = Ar[kc0], a1 = Ar[kc1];
        a.x = ok0 ? a0 : 0.f;
        a.y = ok1 ? a1 : 0.f;

        const float* W0 = W + (size_t)kc0 * N + col0 + lr;
        const float* W1 = W + (size_t)kc1 * N + col0 + lr;
        v2f b0, b1, b2, b3;
        float t00 = W0[0],  t10 = W1[0];
        float t01 = W0[16], t11 = W1[16];
        float t02 = W0[32], t12 = W1[32];
        float t03 = W0[48], t13 = W1[48];
        b0.x = ok0 ? t00 : 0.f;  b0.y = ok1 ? t10 : 0.f;
        b1.x = ok0 ? t01 : 0.f;  b1.y = ok1 ? t11 : 0.f;
        b2.x = ok0 ? t02 : 0.f;  b2.y = ok1 ? t12 : 0.f;
        b3.x = ok0 ? t03 : 0.f;  b3.y = ok1 ? t13 : 0.f;

        acc0 = __builtin_amdgcn_wmma_f32_16x16x4_f32(false, a, false, b0, (short)0, acc0, false, false);
        acc1 = __builtin_amdgcn_wmma_f32_16x16x4_f32(false, a, false, b1, (short)0, acc1, false, false);
        acc2 = __builtin_amdgcn_wmma_f32_16x16x4_f32(false, a, false, b2, (short)0, acc2, false, false);
        acc3 = __builtin_amdgcn_wmma_f32_16x16x4_f32(false, a, false, b3, (short)0, acc3, false, false);
    }

    // Epilogue. C/D layout: VGPR r -> M = row0 + r + 8*hh, N = col0 + t*16 + lr
    float* __restrict__ Crow = C + (size_t)(row0 + 8 * hh) * N + col0 + lr;
    #pragma unroll
    for (int t = 0; t < 4; ++t) {
        v8f acc = (t == 0) ? acc0 : (t == 1) ? acc1 : (t == 2) ? acc2 : acc3;
        const float bv = bias ? bias[col0 + t * 16 + lr] : 0.f;
        #pragma unroll
        for (int r = 0; r < 8; ++r) {
            float v = acc[r] + bv;
            if (do_relu) v = fmaxf(v, 0.f);
            Crow[(size_t)r * N + t * 16] = v;
        }
    }
}

// =====================================================================
// Degree / norm kernels (shared by both graphs & layers)
// =====================================================================
__global__ void deg_init_kernel(float* __restrict__ norm, int n) {
    int i = blockIdx.x * blockDim.x + threadIdx.x;
    if (i < n) norm[i] = 1.0f;   // self-loop fill_value = 1
}

__global__ void deg_add_kernel(float* __restrict__ norm,
                               const int* __restrict__ dst,
                               const float* __restrict__ ew, int e) {
    int i = blockIdx.x * blockDim.x + threadIdx.x;
    if (i < e) atomicAdd(&norm[dst[i]], ew[i]);
}

__global__ void deg_fin_kernel(float* __restrict__ norm, int n) {
    int i = blockIdx.x * blockDim.x + threadIdx.x;
    if (i < n) norm[i] = rsqrtf(norm[i]);
}

// =====================================================================
// agg = h * norm^2 + bias   (self-loop term + conv bias)
// =====================================================================
__global__ void init_agg_kernel(const float* __restrict__ h,
                                const float* __restrict__ norm,
                                const float* __restrict__ bias,
                                float* __restrict__ agg, int rows, int D) {
    size_t idx = (size_t)blockIdx.x * blockDim.x + threadIdx.x;
    if (idx >= (size_t)rows * D) return;
    int p = (int)(idx / D), d = (int)(idx % D);
    float n = norm[p];
    agg[idx] = h[idx] * n * n + bias[d];
}

// s += h * norm^2 + bias   (accumulating layer-2 self term across graphs)
__global__ void s_self_kernel(const float* __restrict__ h,
                              const float* __restrict__ norm,
                              const float* __restrict__ bias,
                              float* __restrict__ s, int rows, int D) {
    size_t idx = (size_t)blockIdx.x * blockDim.x + threadIdx.x;
    if (idx >= (size_t)rows * D) return;
    int p = (int)(idx / D), d = (int)(idx % D);
    float n = norm[p];
    s[idx] += h[idx] * n * n + bias[d];
}

// =====================================================================
// Edge scatter: agg[dst] += h[src] * (ew * norm[src] * norm[dst])
// One block per edge; block covers the feature dim; atomics into L2.
// =====================================================================
__global__ void edge_scatter_kernel(const float* __restrict__ h,
                                    float* __restrict__ agg,
                                    const int* __restrict__ src,
                                    const int* __restrict__ dst,
                                    const float* __restrict__ ew,
                                    const float* __restrict__ norm,
                                    int E, int D) {
    int e = blockIdx.x;
    if (e >= E) return;
    int sp = src[e], dp = dst[e];
    if (e + 1 < E)  // hint next edge's source row toward the caches
        __builtin_prefetch(h + (size_t)src[e + 1] * D, 0, 0);
    float coef = ew[e] * norm[sp] * norm[dp];
    const float* hs = h + (size_t)sp * D;
    float* ad = agg + (size_t)dp * D;
    for (int i = threadIdx.x; i < D; i += blockDim.x)
        atomicAdd(&ad[i], hs[i] * coef);
}

// =====================================================================
// Simple elementwise kernels
// =====================================================================
__global__ void zero_kernel(float* __restrict__ p, size_t n) {
    size_t i = (size_t)blockIdx.x * blockDim.x + threadIdx.x;
    if (i < n) p[i] = 0.f;
}

__global__ void relu_ip_kernel(float* __restrict__ p, size_t n) {
    size_t i = (size_t)blockIdx.x * blockDim.x + threadIdx.x;
    if (i < n) p[i] = fmaxf(p[i], 0.f);
}

// =====================================================================
// Gene -> per-root segment sum: feat[b, r*64+d] = sum_{p: root==r, mask} s[path, d]
// One block per batch element; thread d owns every LDS slot == d (mod 64),
// so no synchronization is needed.
// =====================================================================
__global__ __launch_bounds__(64)
void gene_feat_kernel(const float* __restrict__ s,
                      const int* __restrict__ ids,
                      const int* __restrict__ paths,
                      const unsigned char* __restrict__ mask,
                      const int* __restrict__ root,
                      float* __restrict__ feat) {
    __shared__ float tile[R_ROOT * OUT_DIM];   // 28*64 = 1792 floats
    const int b = blockIdx.x;
    const int d = threadIdx.x;
    for (int i = d; i < R_ROOT * OUT_DIM; i += 64) tile[i] = 0.f;
    const int gid = ids[b];
    const int* gp = paths + (size_t)gid * MP_PATH;
    const unsigned char* gm = mask + (size_t)gid * MP_PATH;
    for (int p = 0; p < MP_PATH; ++p) {
        if (gm[p]) {
            int path = gp[p];
            int r = root[path];
            tile[r * OUT_DIM + d] += s[(size_t)path * OUT_DIM + d];
        }
    }
    for (int i = d; i < R_ROOT * OUT_DIM; i += 64)
        feat[(size_t)b * (R_ROOT * OUT_DIM) + i] = tile[i];
}

// =====================================================================
// BatchNorm (training stats, biased var): per-column mean / rsqrt(var+eps)
// =====================================================================
__global__ __launch_bounds__(256)
void bn_stats_kernel(const float* __restrict__ x,
                     float* __restrict__ mean, float* __restrict__ rstd,
                     int M, int N) {
    __shared__ float sh_s[256];
    __shared__ float sh_q[256];
    const int n = blockIdx.x;
    float s = 0.f, q = 0.f;
    for (int m = threadIdx.x; m < M; m += 256) {
        float v = x[(size_t)m * N + n];
        s += v; q += v * v;
    }
    sh_s[threadIdx.x] = s; sh_q[threadIdx.x] = q;
    __syncthreads();
    for (int off = 128; off > 0; off >>= 1) {
        if (threadIdx.x < off) {
            sh_s[threadIdx.x] += sh_s[threadIdx.x + off];
            sh_q[threadIdx.x] += sh_q[threadIdx.x + off];
        }
        __syncthreads();
    }
    if (threadIdx.x == 0) {
        float mu  = sh_s[0] / (float)M;
        float var = sh_q[0] / (float)M - mu * mu;
        mean[n] = mu;
        rstd[n] = rsqrtf(var + 1e-5f);
    }
}

__global__ void bn_apply_kernel(const float* __restrict__ x,
                                const float* __restrict__ mean,
                                const float* __restrict__ rstd,
                                const float* __restrict__ g,
                                const float* __restrict__ b,
                                float* __restrict__ out,
                                int M, int N, int ldo, int do_relu) {
    size_t idx = (size_t)blockIdx.x * blockDim.x + threadIdx.x;
    if (idx >= (size_t)M * N) return;
    int m = (int)(idx / N), n = (int)(idx % N);
    float v = g[n] * (x[idx] - mean[n]) * rstd[n] + b[n];
    if (do_relu) v = fmaxf(v, 0.f);
    out[(size_t)m * ldo + n] = v;
}

// =====================================================================
// Final: out[b] = sum_d relu(h[b,d]) * w[d] + bias
// =====================================================================
__global__ __launch_bounds__(128)
void final_dot_kernel(const float* __restrict__ h,
                      const float* __restrict__ w,
                      const float* __restrict__ b,
                      float* __restrict__ out) {
    __shared__ float sh[128];
    const int row = blockIdx.x;
    float v = fmaxf(h[(size_t)row * GD_DIM + threadIdx.x], 0.f) * w[threadIdx.x];
    sh[threadIdx.x] = v;
    __syncthreads();
    for (int off = 64; off > 0; off >>= 1) {
        if (threadIdx.x < off) sh[threadIdx.x] += sh[threadIdx.x + off];
        __syncthreads();
    }
    if (threadIdx.x == 0) out[row] = sh[0] + b[0];
}

// =====================================================================
// Host side
// =====================================================================
static inline void gemm(hipStream_t st, const float* A, const float* W,
                        const float* bias, float* C, int M, int N, int K,
                        int do_relu) {
    const int T = (M / 16) * (N / 64);
    const int nblocks = (T + 3) / 4;
    gemm_wmma_f32_kernel<<<nblocks, 128, 0, st>>>(A, W, bias, C, M, N, K, do_relu);
}

static inline int cdiv(size_t a, int b) { return (int)((a + b - 1) / b); }

extern "C" void kernel_launch(void* const* d_in, const int* in_sizes, int n_in,
                              void* d_out, int out_size, void* d_ws, size_t ws_size,
                              hipStream_t stream) {
    (void)in_sizes; (void)n_in; (void)out_size; (void)ws_size;
    // ---- inputs (setup_inputs order) ----
    const float* x1        = (const float*)d_in[0];
    const float* x2        = (const float*)d_in[1];
    const int*   e_src     = (const int*)d_in[2];
    const int*   e_dst     = (const int*)d_in[3];
    const float* e_w       = (const float*)d_in[4];
    const float* gw1[2]    = { (const float*)d_in[5],  (const float*)d_in[9]  };
    const float* gb1[2]    = { (const float*)d_in[6],  (const float*)d_in[10] };
    const float* gw2[2]    = { (const float*)d_in[7],  (const float*)d_in[11] };
    const float* gb2[2]    = { (const float*)d_in[8],  (const float*)d_in[12] };
    const float* pl1_w     = (const float*)d_in[13];
    const float* pl1_b     = (const float*)d_in[14];
    const float* bn1_g     = (const float*)d_in[15];
    const float* bn1_b     = (const float*)d_in[16];
    const float* pl2_w     = (const float*)d_in[17];
    const float* pl2_b     = (const float*)d_in[18];
    const float* bn2_g     = (const float*)d_in[19];
    const float* bn2_b     = (const float*)d_in[20];
    const float* fc_w      = (const float*)d_in[21];
    const float* fc_b      = (const float*)d_in[22];
    const float* fc1_w     = (const float*)d_in[23];
    const float* fc1_b     = (const float*)d_in[24];
    const float* fc2_w     = (const float*)d_in[25];
    const float* fc2_b     = (const float*)d_in[26];
    const int*   root      = (const int*)d_in[27];
    const int*   gpaths    = (const int*)d_in[28];
    const unsigned char* gmask = (const unsigned char*)d_in[29];
    const int*   head_ids  = (const int*)d_in[30];
    const int*   tail_ids  = (const int*)d_in[31];
    float* out = (float*)d_out;

    // ---- workspace layout (floats) ----
    float* ws = (float*)d_ws;
    size_t off = 0;
    float* norm = ws + off; off += 32768;                        // P
    float* h1   = ws + off; off += (size_t)P_NODES * H_DIM;      // 30000*256
    float* a1   = ws + off; off += (size_t)P_NODES * H_DIM;
    float* h2   = ws + off; off += (size_t)P_NODES * OUT_DIM;    // 30000*64
    float* sbuf = ws + off; off += (size_t)P_NODES * OUT_DIM;
    float* feat = ws + off; off += (size_t)B_BATCH * R_ROOT * OUT_DIM; // 4096*1792
    float* y1   = ws + off; off += (size_t)B_BATCH * 256;
    float* y2   = ws + off; off += (size_t)B_BATCH * GD_DIM;
    float* catb = ws + off; off += (size_t)B_BATCH * 2 * GD_DIM;
    float* h3   = ws + off; off += (size_t)B_BATCH * 256;
    float* h4   = ws + off; off += (size_t)B_BATCH * GD_DIM;
    float* bnmu = ws + off; off += 256;
    float* bnrs = ws + off; off += 256;

    // ---- 1) symmetric-norm degrees (shared by both graphs / layers) ----
    deg_init_kernel<<<cdiv(P_NODES, 256), 256, 0, stream>>>(norm, P_NODES);
    deg_add_kernel<<<cdiv(E_EDGES, 256), 256, 0, stream>>>(norm, e_dst, e_w, E_EDGES);
    deg_fin_kernel<<<cdiv(P_NODES, 256), 256, 0, stream>>>(norm, P_NODES);

    // ---- 2) s accumulator = out1 + out2 ----
    zero_kernel<<<cdiv((size_t)P_NODES * OUT_DIM, 256), 256, 0, stream>>>(sbuf, (size_t)P_NODES * OUT_DIM);

    // ---- 3) per-graph 2-layer GCN ----
    for (int g = 0; g < 2; ++g) {
        const float* x = g ? x2 : x1;
        // layer 1: h1 = x @ w1
        gemm(stream, x, gw1[g], nullptr, h1, P_NODES, H_DIM, IN_DIM, 0);
        // a1 = h1*norm^2 + b1 ; scatter edges ; relu
        init_agg_kernel<<<cdiv((size_t)P_NODES * H_DIM, 256), 256, 0, stream>>>(h1, norm, gb1[g], a1, P_NODES, H_DIM);
        edge_scatter_kernel<<<E_EDGES, 256, 0, stream>>>(h1, a1, e_src, e_dst, e_w, norm, E_EDGES, H_DIM);
        relu_ip_kernel<<<cdiv((size_t)P_NODES * H_DIM, 256), 256, 0, stream>>>(a1, (size_t)P_NODES * H_DIM);
        // layer 2: h2 = relu(a1) @ w2 ; accumulate into s
        gemm(stream, a1, gw2[g], nullptr, h2, P_NODES, OUT_DIM, H_DIM, 0);
        s_self_kernel<<<cdiv((size_t)P_NODES * OUT_DIM, 256), 256, 0, stream>>>(h2, norm, gb2[g], sbuf, P_NODES, OUT_DIM);
        edge_scatter_kernel<<<E_EDGES, 64, 0, stream>>>(h2, sbuf, e_src, e_dst, e_w, norm, E_EDGES, OUT_DIM);
    }

    // ---- 4) fuse(head) -> cat[:, :128], fuse(tail) -> cat[:, 128:] ----
    for (int t = 0; t < 2; ++t) {
        const int* ids = t ? tail_ids : head_ids;
        float* catp = catb + (t ? GD_DIM : 0);
        gene_feat_kernel<<<B_BATCH, 64, 0, stream>>>(sbuf, ids, gpaths, gmask, root, feat);
        gemm(stream, feat, pl1_w, pl1_b, y1, B_BATCH, 256, R_ROOT * OUT_DIM, 0);
        bn_stats_kernel<<<256, 256, 0, stream>>>(y1, bnmu, bnrs, B_BATCH, 256);
        bn_apply_kernel<<<cdiv((size_t)B_BATCH * 256, 256), 256, 0, stream>>>(
            y1, bnmu, bnrs, bn1_g, bn1_b, y1, B_BATCH, 256, 256, 1);
        gemm(stream, y1, pl2_w, pl2_b, y2, B_BATCH, GD_DIM, 256, 0);
        bn_stats_kernel<<<GD_DIM, 256, 0, stream>>>(y2, bnmu, bnrs, B_BATCH, GD_DIM);
        bn_apply_kernel<<<cdiv((size_t)B_BATCH * GD_DIM, 256), 256, 0, stream>>>(
            y2, bnmu, bnrs, bn2_g, bn2_b, catp, B_BATCH, GD_DIM, 2 * GD_DIM, 0);
    }

    // ---- 5) final MLP ----
    gemm(stream, catb, fc_w, fc_b, h3, B_BATCH, 256, 2 * GD_DIM, 1);   // relu(cat@fc_w+fc_b)
    gemm(stream, h3, fc1_w, fc1_b, h4, B_BATCH, GD_DIM, 256, 0);       // final (no relu yet)
    final_dot_kernel<<<B_BATCH, 128, 0, stream>>>(h4, fc2_w, fc2_b, out); // relu + dot + bias
}